// DiffTexture_88845693485816
// MI455X (gfx1250) — compile-verified
//
#include <hip/hip_runtime.h>

// Bilinear texture sampling for MI455X (gfx1250, wave32).
// Gather-bound: texture (48 MB) is L2-resident (192 MB L2); uv/output streams
// use NT hints so they don't evict it. Two samples per iteration from one
// b128 uv load -> 8 b96 gathers in flight per loadcnt wait, merged 24B store.

typedef float v4f __attribute__((ext_vector_type(4)));

#define TEX_W 2048
#define TEX_H 2048

__device__ __forceinline__ void sample_one(
    const float* __restrict__ tex, float u_in, float v_in,
    float& rx, float& ry, float& rz)
{
    // map [-1,1] -> [0, dim-1]
    float u = (u_in + 1.0f) * (0.5f * (float)(TEX_W - 1));
    float v = (v_in + 1.0f) * (0.5f * (float)(TEX_H - 1));

    float fu0 = floorf(u);
    float fv0 = floorf(v);
    int u0 = (int)fu0;
    int v0 = (int)fv0;
    int u1 = (int)ceilf(u);
    int v1 = (int)ceilf(v);
    float a = u - fu0;
    float b = v - fv0;

    const float* __restrict__ p00 = tex + (u0 * TEX_H + v0) * 3;
    const float* __restrict__ p10 = tex + (u1 * TEX_H + v0) * 3;
    const float* __restrict__ p01 = tex + (u0 * TEX_H + v1) * 3;
    const float* __restrict__ p11 = tex + (u1 * TEX_H + v1) * 3;

    // 12B contiguous gathers -> global_load_b96 (regular temporal: L2-resident)
    float c00x = p00[0], c00y = p00[1], c00z = p00[2];
    float c10x = p10[0], c10y = p10[1], c10z = p10[2];
    float c01x = p01[0], c01y = p01[1], c01z = p01[2];
    float c11x = p11[0], c11y = p11[1], c11z = p11[2];

    float ia = 1.0f - a;
    float ib = 1.0f - b;

    // (T[u0,v0]*a + T[u1,v0]*(1-a))*b + (T[u0,v1]*a + T[u1,v1]*(1-a))*(1-b)
    rx = fmaf(fmaf(c00x, a, c10x * ia), b, fmaf(c01x, a, c11x * ia) * ib);
    ry = fmaf(fmaf(c00y, a, c10y * ia), b, fmaf(c01y, a, c11y * ia) * ib);
    rz = fmaf(fmaf(c00z, a, c10z * ia), b, fmaf(c01z, a, c11z * ia) * ib);
}

__global__ __launch_bounds__(256, 1) void DiffTexture_bilinear_kernel(
    const v4f* __restrict__ uv2,   // two uv pairs per element
    const float* __restrict__ tex,
    float* __restrict__ out,
    int n)
{
    const int stride = gridDim.x * blockDim.x;
    const int gid = blockIdx.x * blockDim.x + threadIdx.x;
    const int npairs = n >> 1;

    for (int p = gid; p < npairs; p += stride) {
        // Prefetch next grid-stride iteration's uv line into all cache levels
        // (same wave consumes it next iteration) -> global_prefetch_b8
        __builtin_prefetch((const void*)(uv2 + p + stride), 0, 3);

        // One-touch stream: NT load keeps L2 for the texture working set.
        v4f uv = __builtin_nontemporal_load(uv2 + p);

        float r0x, r0y, r0z, r1x, r1y, r1z;
        sample_one(tex, uv.x, uv.y, r0x, r0y, r0z);
        sample_one(tex, uv.z, uv.w, r1x, r1y, r1z);

        // 24B contiguous one-touch output: NT stores, mergeable to b128+b64.
        float* o = out + (size_t)p * 6u;
        __builtin_nontemporal_store(r0x, o + 0);
        __builtin_nontemporal_store(r0y, o + 1);
        __builtin_nontemporal_store(r0z, o + 2);
        __builtin_nontemporal_store(r1x, o + 3);
        __builtin_nontemporal_store(r1y, o + 4);
        __builtin_nontemporal_store(r1z, o + 5);
    }

    // Tail for odd n (not hit for N=4194304, kept for correctness).
    if (gid == 0 && (n & 1)) {
        const float* uvt = (const float*)uv2 + (size_t)(n - 1) * 2u;
        float rx, ry, rz;
        sample_one(tex, uvt[0], uvt[1], rx, ry, rz);
        float* o = out + (size_t)(n - 1) * 3u;
        __builtin_nontemporal_store(rx, o + 0);
        __builtin_nontemporal_store(ry, o + 1);
        __builtin_nontemporal_store(rz, o + 2);
    }
}

extern "C" void kernel_launch(void* const* d_in, const int* in_sizes, int n_in,
                              void* d_out, int out_size, void* d_ws, size_t ws_size,
                              hipStream_t stream) {
    const v4f*  uv2 = (const v4f*)d_in[0];    // [N,2] f32, read as [N/2,4]
    const float* tex = (const float*)d_in[1]; // [2048,2048,3] f32
    float* out = (float*)d_out;               // [N,3] f32

    int n = in_sizes[0] / 2;                  // number of samples

    // 2048 blocks x 256 threads (8 wave32/block); each thread iterates
    // npairs/524288 = 4x for N=4M -> prefetch target + cross-sample MLP.
    int npairs = n >> 1;
    int blocks = (npairs + 255) / 256;
    if (blocks > 2048) blocks = 2048;
    if (blocks < 1) blocks = 1;

    DiffTexture_bilinear_kernel<<<blocks, 256, 0, stream>>>(uv2, tex, out, n);
}